// GraphPool_10110353015351
// MI455X (gfx1250) — compile-verified
//
#include <hip/hip_runtime.h>
#include <hip/hip_bf16.h>

typedef __attribute__((ext_vector_type(2))) float v2f;
typedef __attribute__((ext_vector_type(8))) float v8f;

#define N_FULL 16384
#define N_KEEP 8192
#define DIM    128

// ---------------------------------------------------------------------------
// Kernel 1: scores = sigmoid((X @ W) / ||W||) via V_WMMA_F32_16X16X4_F32.
// One wave handles a 16-row tile of X; K looped 128/4 = 32 WMMA steps.
// A layout (16x4 f32): lanes 0-15 hold M=lane, VGPR0=K0,VGPR1=K1;
//                      lanes 16-31 hold M=lane-16, VGPR0=K2,VGPR1=K3.
// B (4x16) mirrors: VGPR0 = rows K0/K2 across lane halves, VGPR1 = K1/K3;
// every column of B is W[k], so the per-lane value depends only on K.
// D (16x16 f32): lanes 0-15 carry M=vgpr (N=lane), lanes 16-31 carry M=vgpr+8.
// Also emits the 64-bit sort key (score_bits<<32)|~row for the top-k sort.
// ---------------------------------------------------------------------------
__global__ __launch_bounds__(256) void k_scores(
    const float* __restrict__ X, const float* __restrict__ W,
    float* __restrict__ scores, float* __restrict__ Wout,
    unsigned long long* __restrict__ keys) {
  const int tid  = threadIdx.x;
  const int lane = tid & 31;
  const int wave = tid >> 5;
  const int base = (blockIdx.x * 8 + wave) * 16;   // 16 rows per wave

  // ||W||: redundant per-thread scalar loop over 128 elems (negligible).
  float nrm = 0.f;
  #pragma unroll 8
  for (int k = 0; k < DIM; ++k) { float w = W[k]; nrm += w * w; }
  const float invn = 1.0f / sqrtf(nrm);

  const int m    = lane & 15;
  const int half = lane >> 4;          // 0: lanes 0-15, 1: lanes 16-31
  const int row  = base + m;
  const float* xr = X + (size_t)row * DIM;

  v8f c = {};
  for (int k0 = 0; k0 < DIM; k0 += 4) {
    const int kk = k0 + 2 * half;
    v2f a, b;
    a.x = xr[kk];  a.y = xr[kk + 1];
    b.x = W[kk];   b.y = W[kk + 1];
    c = __builtin_amdgcn_wmma_f32_16x16x4_f32(
        /*neg_a=*/false, a, /*neg_b=*/false, b,
        /*c_mod=*/(short)0, c, /*reuse_a=*/false, /*reuse_b=*/false);
  }

  // Extract column N=0 of D: lane 0 holds rows base+0..7, lane 16 rows base+8..15.
  if (m == 0) {
    #pragma unroll
    for (int q = 0; q < 8; ++q) {
      const int r = base + half * 8 + q;
      float s = c[q] * invn;
      s = 1.0f / (1.0f + __expf(-s));            // sigmoid, result in (0,1)
      scores[r] = s;
      const unsigned int sb = __float_as_uint(s); // positive float: bits monotone
      keys[r] = ((unsigned long long)sb << 32) |
                (unsigned int)(~(unsigned int)r); // ~r => stable ties (low idx first)
    }
  }

  // Pass-through output: W
  if (blockIdx.x == 0 && tid < DIM) Wout[tid] = W[tid];
}

// ---------------------------------------------------------------------------
// Kernel 2: full bitonic sort (descending) of 16384 u64 keys in 128 KB LDS.
// Single workgroup of 1024 threads (32 waves on one WGP; CDNA5 LDS = 320 KB).
// Then emit idx (as float), int row indices and float values for top 8192.
// ---------------------------------------------------------------------------
__global__ __launch_bounds__(1024) void k_sort(
    unsigned long long* __restrict__ keys,
    int* __restrict__ ridx, float* __restrict__ rval,
    float* __restrict__ idx_out) {
  __shared__ unsigned long long sk[N_FULL];      // 131072 bytes
  const int tid = threadIdx.x;

  for (int i = tid; i < N_FULL; i += 1024) sk[i] = keys[i];
  __syncthreads();

  for (unsigned k = 2; k <= (unsigned)N_FULL; k <<= 1) {
    for (unsigned j = k >> 1; j > 0; j >>= 1) {
      for (unsigned i = tid; i < (unsigned)N_FULL; i += 1024) {
        const unsigned ix = i ^ j;
        if (ix > i) {
          const unsigned long long a = sk[i];
          const unsigned long long b = sk[ix];
          const bool up = ((i & k) == 0);
          if (up ? (a < b) : (a > b)) { sk[i] = b; sk[ix] = a; }
        }
      }
      __syncthreads();
    }
  }

  for (int i = tid; i < N_FULL; i += 1024) keys[i] = sk[i];
  for (int i = tid; i < N_KEEP; i += 1024) {
    const unsigned long long kv = sk[i];
    const unsigned int r = ~(unsigned int)(kv & 0xffffffffu);
    ridx[i]    = (int)r;
    rval[i]    = __uint_as_float((unsigned int)(kv >> 32));
    idx_out[i] = (float)r;
  }
}

// ---------------------------------------------------------------------------
// Kernel 3: new_X[i,:] = X[idx[i],:] * values[i]. One wave per row, float4.
// ---------------------------------------------------------------------------
__global__ __launch_bounds__(256) void k_newx(
    const float* __restrict__ X, const int* __restrict__ ridx,
    const float* __restrict__ rval, float* __restrict__ newX) {
  const int wave = threadIdx.x >> 5;
  const int lane = threadIdx.x & 31;
  const int i = blockIdx.x * 8 + wave;           // grid 1024 -> 8192 rows
  const int r = ridx[i];
  const float v = rval[i];
  const float4 x = ((const float4*)(X + (size_t)r * DIM))[lane];
  float4 o; o.x = x.x * v; o.y = x.y * v; o.z = x.z * v; o.w = x.w * v;
  ((float4*)(newX + (size_t)i * DIM))[lane] = o;
}

// ---------------------------------------------------------------------------
// Kernel 4: A_new[i][j] = A[idx[i]][idx[j]]. One block per output row.
// Column indices staged once per block in 32 KB LDS; source row (64 KB)
// prefetched via global_prefetch_b8; stores fully coalesced.
// ---------------------------------------------------------------------------
__global__ __launch_bounds__(256) void k_gather(
    const float* __restrict__ A, const int* __restrict__ ridx,
    float* __restrict__ Aout) {
  __shared__ int cidx[N_KEEP];                   // 32 KB
  const int tid = threadIdx.x;
  const int i = blockIdx.x;
  const int r = ridx[i];
  const float* arow = A + (size_t)r * N_FULL;

  // Prefetch the full 64 KB source row (256 threads x 2 x 128B-line stride).
  __builtin_prefetch(arow + tid * 64, 0, 3);       // +tid*256 bytes
  __builtin_prefetch(arow + tid * 64 + 32, 0, 3);  // +128 bytes

  for (int j = tid; j < N_KEEP; j += 256) cidx[j] = ridx[j];
  __syncthreads();

  float* orow = Aout + (size_t)i * N_KEEP;
  #pragma unroll 4
  for (int j = tid; j < N_KEEP; j += 256) {
    orow[j] = arow[cidx[j]];
  }
}

// ---------------------------------------------------------------------------
extern "C" void kernel_launch(void* const* d_in, const int* in_sizes, int n_in,
                              void* d_out, int out_size, void* d_ws, size_t ws_size,
                              hipStream_t stream) {
  const float* A = (const float*)d_in[0];   // [16384,16384]
  const float* X = (const float*)d_in[1];   // [16384,128]
  const float* W = (const float*)d_in[2];   // [1,128]

  float* out    = (float*)d_out;
  float* A_new  = out;                                   // 8192*8192
  float* newX   = out + (size_t)N_KEEP * N_KEEP;         // 8192*128
  float* idxf   = newX + (size_t)N_KEEP * DIM;           // 8192
  float* scores = idxf + N_KEEP;                         // 16384
  float* Wout   = scores + N_FULL;                       // 128

  unsigned long long* keys = (unsigned long long*)d_ws;            // 128 KB
  int*   ridx = (int*)((char*)d_ws + 131072);                      // 32 KB
  float* rval = (float*)((char*)d_ws + 131072 + 32768);            // 32 KB

  k_scores<<<128,  256, 0, stream>>>(X, W, scores, Wout, keys);
  k_sort  <<<1,   1024, 0, stream>>>(keys, ridx, rval, idxf);
  k_newx  <<<1024, 256, 0, stream>>>(X, ridx, rval, newX);
  k_gather<<<8192, 256, 0, stream>>>(A, ridx, A_new);
}